// Aggregator_13048110645350
// MI455X (gfx1250) — compile-verified
//
#include <hip/hip_runtime.h>
#include <hip/hip_bf16.h>
#include <math.h>

typedef __attribute__((ext_vector_type(2))) float v2f;
typedef __attribute__((ext_vector_type(8))) float v8f;

#define N_ENT_C   100000
#define CH_C      128
#define N_REL_C   64
#define N_USERS_C 50000
#define N_FACT_C  4

// ---------------------------------------------------------------- zero
__global__ void zero_kernel(float* __restrict__ p, long n) {
  long i = (long)blockIdx.x * blockDim.x + threadIdx.x;
  long stride = (long)gridDim.x * blockDim.x;
  for (; i < n; i += stride) p[i] = 0.0f;
}

// ------------------------------------------------- KG edge scatter-add
// one wave (32 lanes) per edge; lane owns 4 channels
__global__ __launch_bounds__(256) void kg_edge_kernel(
    const float* __restrict__ ent, const int* __restrict__ edge_index,
    const int* __restrict__ edge_type, const float* __restrict__ weight,
    float* __restrict__ entity_agg, float* __restrict__ cnt, int n_edge) {
  int gw = (int)(((long)blockIdx.x * blockDim.x + threadIdx.x) >> 5);
  if (gw >= n_edge) return;
  int lane = threadIdx.x & 31;
  int head = edge_index[gw];
  int tail = edge_index[n_edge + gw];
  int rel  = edge_type[gw] - 1;   // 1-based in reference
  const float4 t = *(const float4*)(ent + (size_t)tail * CH_C + lane * 4);
  const float4 w = *(const float4*)(weight + (size_t)rel * CH_C + lane * 4);
  float* dst = entity_agg + (size_t)head * CH_C + lane * 4;
  atomicAdd(dst + 0, t.x * w.x);
  atomicAdd(dst + 1, t.y * w.y);
  atomicAdd(dst + 2, t.z * w.z);
  atomicAdd(dst + 3, t.w * w.w);
  if (lane == 0) atomicAdd(cnt + head, 1.0f);
}

// ------------------------------------------------- scatter-mean divide
__global__ void kg_norm_kernel(float* __restrict__ agg,
                               const float* __restrict__ cnt, long n) {
  long i = (long)blockIdx.x * blockDim.x + threadIdx.x;
  if (i >= n) return;
  float c = fmaxf(cnt[i >> 7], 1.0f);
  agg[i] = agg[i] / c;
}

// ------------------------------------------------- sparse user @ entity
__global__ __launch_bounds__(256) void user_spmm_kernel(
    const float* __restrict__ ent, const int* __restrict__ rows,
    const int* __restrict__ cols, const float* __restrict__ vals,
    float* __restrict__ user_agg, int nnz) {
  int gw = (int)(((long)blockIdx.x * blockDim.x + threadIdx.x) >> 5);
  if (gw >= nnz) return;
  int lane = threadIdx.x & 31;
  int r = rows[gw];
  int c = cols[gw];
  float v = vals[gw];
  const float4 e = *(const float4*)(ent + (size_t)c * CH_C + lane * 4);
  float* dst = user_agg + (size_t)r * CH_C + lane * 4;
  atomicAdd(dst + 0, v * e.x);
  atomicAdd(dst + 1, v * e.y);
  atomicAdd(dst + 2, v * e.z);
  atomicAdd(dst + 3, v * e.w);
}

// ---------------------------- disen_weight = softmax(att,-1) @ weight
__global__ __launch_bounds__(256) void disen_kernel(
    const float* __restrict__ att, const float* __restrict__ weight,
    float* __restrict__ dw) {
  __shared__ float sm[N_FACT_C][N_REL_C];
  __shared__ float rmax[N_FACT_C], rsum[N_FACT_C];
  int tid = threadIdx.x;
  if (tid < N_FACT_C * N_REL_C) sm[tid >> 6][tid & 63] = att[tid];
  __syncthreads();
  if (tid < N_FACT_C) {
    float m = -INFINITY;
    for (int r = 0; r < N_REL_C; ++r) m = fmaxf(m, sm[tid][r]);
    float s = 0.0f;
    for (int r = 0; r < N_REL_C; ++r) s += __expf(sm[tid][r] - m);
    rmax[tid] = m;
    rsum[tid] = s;
  }
  __syncthreads();
  if (tid < N_FACT_C * N_REL_C) {
    int f = tid >> 6, r = tid & 63;
    sm[f][r] = __expf(sm[f][r] - rmax[f]) / rsum[f];
  }
  __syncthreads();
  if (tid < CH_C) {
    for (int f = 0; f < N_FACT_C; ++f) {
      float acc = 0.0f;
      for (int r = 0; r < N_REL_C; ++r)
        acc = fmaf(sm[f][r], weight[r * CH_C + tid], acc);
      dw[f * CH_C + tid] = acc;
    }
  }
}

// ---------------- WMMA: score = softmax(U @ L^T), user_agg *= 1 + score@dw
// 8 waves per block; each wave owns a 16-user tile.
// f32 WMMA fragments (ISA 7.12.2): A 16x4: lanes 0-15 -> K=0,1 ; lanes 16-31 -> K=2,3
// C/D: VGPR v holds (M = v + 8*(lane/16), N = lane%16)
// B panel is zero-padded to 16 rows in LDS so the inner loop has NO divergent
// branch (removes per-step saveexec/exec-restore around the ds_load).
__global__ __launch_bounds__(256) void user_mix_kernel(
    const float* __restrict__ user_emb, const float* __restrict__ latent_emb,
    const float* __restrict__ dw, float* __restrict__ user_agg, int n_users) {
  __shared__ float latB[16 * CH_C];          // rows 0..3 = latent_emb, 4..15 = 0
  __shared__ float dw_lds[N_FACT_C * CH_C];
  __shared__ float score_lds[8][16][4];

  int tid = threadIdx.x;
  for (int i = tid; i < 16 * CH_C; i += 256)
    latB[i] = (i < N_FACT_C * CH_C) ? latent_emb[i] : 0.0f;
  for (int i = tid; i < N_FACT_C * CH_C; i += 256)
    dw_lds[i] = dw[i];
  __syncthreads();

  int waveId = tid >> 5;
  int lane = tid & 31;
  int half = lane >> 4;   // which K-pair this lane holds
  int lm = lane & 15;     // row (A) / column (B) index
  int u0 = (blockIdx.x * 8 + waveId) * 16;
  int u0r = (u0 < n_users - 16) ? u0 : (n_users - 16);  // clamped for reads
  bool valid = (u0 < n_users);                          // wave-uniform store guard

  // ---- score GEMM: C[16x16] += A(16x4) * B(4x16 padded), 32 K-steps (K=128)
  v8f acc = {0.f, 0.f, 0.f, 0.f, 0.f, 0.f, 0.f, 0.f};
  const float* arow = user_emb + (size_t)(u0r + lm) * CH_C;
  const float* brow = latB + lm * CH_C;  // rows >= 4 are zero-padded
  for (int k = 0; k < CH_C; k += 4) {
    v2f a;
    a.x = arow[k + 2 * half];
    a.y = arow[k + 2 * half + 1];
    v2f b;
    b.x = brow[k + 2 * half];
    b.y = brow[k + 2 * half + 1];
    acc = __builtin_amdgcn_wmma_f32_16x16x4_f32(false, a, false, b,
                                                (short)0, acc, false, false);
  }
  if (lm < N_FACT_C) {
#pragma unroll
    for (int v = 0; v < 8; ++v)
      score_lds[waveId][v + 8 * half][lm] = acc[v];
  }
  __syncthreads();

  // ---- softmax over the 4 factors, one user-row per lane 0..15
  if (lane < 16) {
    float s0 = score_lds[waveId][lane][0];
    float s1 = score_lds[waveId][lane][1];
    float s2 = score_lds[waveId][lane][2];
    float s3 = score_lds[waveId][lane][3];
    float m = fmaxf(fmaxf(s0, s1), fmaxf(s2, s3));
    float e0 = __expf(s0 - m), e1 = __expf(s1 - m);
    float e2 = __expf(s2 - m), e3 = __expf(s3 - m);
    float inv = 1.0f / (e0 + e1 + e2 + e3);
    score_lds[waveId][lane][0] = e0 * inv;
    score_lds[waveId][lane][1] = e1 * inv;
    score_lds[waveId][lane][2] = e2 * inv;
    score_lds[waveId][lane][3] = e3 * inv;
  }
  __syncthreads();

  // ---- mix GEMM: K=4 exactly, one WMMA per 16-wide N tile (8 tiles = 128 ch)
  v2f a2;
  a2.x = score_lds[waveId][lm][2 * half];
  a2.y = score_lds[waveId][lm][2 * half + 1];
  for (int t = 0; t < 8; ++t) {
    int n0 = t * 16;
    v2f b2;
    b2.x = dw_lds[(2 * half) * CH_C + n0 + lm];
    b2.y = dw_lds[(2 * half + 1) * CH_C + n0 + lm];
    v8f macc = {0.f, 0.f, 0.f, 0.f, 0.f, 0.f, 0.f, 0.f};
    macc = __builtin_amdgcn_wmma_f32_16x16x4_f32(false, a2, false, b2,
                                                 (short)0, macc, false, false);
    if (valid) {
#pragma unroll
      for (int v = 0; v < 8; ++v) {
        int m = v + 8 * half;
        size_t idx = (size_t)(u0 + m) * CH_C + n0 + lm;
        float ua = user_agg[idx];
        user_agg[idx] = fmaf(ua, macc[v], ua);  // ua*mix + ua
      }
    }
  }
}

// ---------------------------------------------------------------- launch
extern "C" void kernel_launch(void* const* d_in, const int* in_sizes, int n_in,
                              void* d_out, int out_size, void* d_ws, size_t ws_size,
                              hipStream_t stream) {
  const float* entity_emb = (const float*)d_in[0];
  const float* user_emb   = (const float*)d_in[1];
  const float* latent_emb = (const float*)d_in[2];
  const int*   edge_index = (const int*)d_in[3];
  const int*   edge_type  = (const int*)d_in[4];
  const float* weight     = (const float*)d_in[5];
  const float* disen_att  = (const float*)d_in[6];
  const int*   irows      = (const int*)d_in[7];
  const int*   icols      = (const int*)d_in[8];
  const float* ivals      = (const float*)d_in[9];

  int n_edge = in_sizes[4];  // edge_type length
  int nnz    = in_sizes[9];  // interact_vals length

  float* entity_agg = (float*)d_out;                          // [N_ENT*CH]
  float* user_agg   = (float*)d_out + (size_t)N_ENT_C * CH_C; // [N_USERS*CH]
  float* cnt        = (float*)d_ws;                           // [N_ENT]
  float* dw         = (float*)d_ws + 100352;                  // [N_FACT*CH], aligned past cnt

  // 1. zero outputs + counts (determinism across graph replays)
  long zero_total = (long)N_ENT_C * CH_C + (long)N_USERS_C * CH_C;
  zero_kernel<<<4096, 256, 0, stream>>>((float*)d_out, zero_total);
  zero_kernel<<<256, 256, 0, stream>>>(cnt, (long)N_ENT_C);

  // 2. KG edge scatter-add (wave per edge)
  {
    long threads = (long)n_edge * 32;
    int blocks = (int)((threads + 255) / 256);
    kg_edge_kernel<<<blocks, 256, 0, stream>>>(entity_emb, edge_index, edge_type,
                                               weight, entity_agg, cnt, n_edge);
  }
  // 3. scatter-mean normalize
  {
    long n = (long)N_ENT_C * CH_C;
    kg_norm_kernel<<<(int)((n + 255) / 256), 256, 0, stream>>>(entity_agg, cnt, n);
  }
  // 4. sparse user aggregation (wave per nnz)
  {
    long threads = (long)nnz * 32;
    int blocks = (int)((threads + 255) / 256);
    user_spmm_kernel<<<blocks, 256, 0, stream>>>(entity_emb, irows, icols, ivals,
                                                 user_agg, nnz);
  }
  // 5. disen_weight = softmax(att) @ weight
  disen_kernel<<<1, 256, 0, stream>>>(disen_att, weight, dw);

  // 6. WMMA score/softmax/mix + in-place user_agg update
  {
    int tiles = (N_USERS_C + 15) / 16;   // 3125
    int blocks = (tiles + 7) / 8;        // 391
    user_mix_kernel<<<blocks, 256, 0, stream>>>(user_emb, latent_emb, dw,
                                                user_agg, N_USERS_C);
  }
}